// ROIAlignRotated3D_61151744360591
// MI455X (gfx1250) — compile-verified
//
#include <hip/hip_runtime.h>

// ROI-align-rotated-3D for MI455X (gfx1250), z-interp+z-pool as V_WMMA_F32_16X16X4_F32.
// Shapes fixed by reference: B=2, C=128, W=L=H=64, OUT=7^3, SR=2, N from in_sizes.

typedef float v2f __attribute__((ext_vector_type(2)));
typedef float v8f __attribute__((ext_vector_type(8)));

#define C_CH 128
#define WDIM 64
#define LDIM 64
#define HDIM 64

__device__ __forceinline__ float sfirst(float v) {
    return __uint_as_float(__builtin_amdgcn_readfirstlane(__float_as_uint(v)));
}
__device__ __forceinline__ int sfirst(int v) {
    return __builtin_amdgcn_readfirstlane(v);
}

__global__ __launch_bounds__(256)
void roi_align_rot3d_wmma(const float* __restrict__ in,
                          const float* __restrict__ rois,
                          const float* __restrict__ scale_p,
                          float* __restrict__ out)
{
    __shared__ float sZc[16][64];   // [z-bin padded to 16][z] exact z operator (weights*valid*0.5)
    __shared__ v2f   sB[16][32];    // [k-chunk][lane] pre-swizzled WMMA B operand
    __shared__ int   sSpat[16];     // 16 corner spatial offsets: (xi<<12)|(yi<<6)
    __shared__ float sW[16];        // 16 corner weights: valid*0.25*wx*wy
    __shared__ int   sMeta[3];      // batch, ckmin, ckmax

    const int bin  = blockIdx.x;          // 0..48
    const int roi  = blockIdx.y;
    const int bx   = bin / 7, by = bin % 7;
    const int tid  = threadIdx.x;
    const int lane = tid & 31;
    const int wave = tid >> 5;            // 8 waves -> 8 groups of 16 channels

    for (int i = tid; i < 16 * 64; i += 256) ((float*)sZc)[i] = 0.0f;
    __syncthreads();

    if (tid == 0) {
        const float scale = scale_p[0];
        const float* r = rois + roi * 8;
        const int   b  = (int)r[0];
        const float cx = r[1] * scale, cy = r[2] * scale, cz = r[3] * scale;
        const float sx = r[4] * scale, sy = r[5] * scale, sz = r[6] * scale;
        const float ct = cosf(r[7]), st = sinf(r[7]);

        // ---- exact z operator: trilinear z weights * z-validity * (1/2 pool) ----
        int zmin = 1 << 30, zmax = -1;
        for (int zs = 0; zs < 14; ++zs) {
            float z = -0.5f * sz + ((float)zs + 0.5f) * (sz * (1.0f / 14.0f)) + cz;
            if (z > -1.0f && z < (float)HDIM) {
                float zc = fminf(fmaxf(z, 0.0f), (float)(HDIM - 1));
                int z0 = (int)floorf(zc);
                int z1 = min(z0 + 1, HDIM - 1);
                float lz = zc - (float)z0;
                sZc[zs >> 1][z0] += 0.5f * (1.0f - lz);
                sZc[zs >> 1][z1] += 0.5f * lz;
                zmin = min(zmin, z0); zmax = max(zmax, z1);
            }
        }
        sMeta[0] = b;
        sMeta[1] = (zmax < 0) ? 1 : (zmin >> 2);
        sMeta[2] = (zmax < 0) ? 0 : (zmax >> 2);

        // ---- 4 xy supersamples of this bin -> 16 (corner, weight) entries ----
        for (int s4 = 0; s4 < 4; ++s4) {
            int xs = 2 * bx + (s4 & 1);
            int ys = 2 * by + (s4 >> 1);
            float xo = -0.5f * sx + ((float)xs + 0.5f) * (sx * (1.0f / 14.0f));
            float yo = -0.5f * sy + ((float)ys + 0.5f) * (sy * (1.0f / 14.0f));
            float x = ct * xo - st * yo + cx;
            float y = st * xo + ct * yo + cy;
            bool v = (x > -1.0f) && (x < (float)WDIM) && (y > -1.0f) && (y < (float)LDIM);
            float xc = fminf(fmaxf(x, 0.0f), (float)(WDIM - 1));
            float yc = fminf(fmaxf(y, 0.0f), (float)(LDIM - 1));
            int x0 = (int)floorf(xc), y0 = (int)floorf(yc);
            int x1 = min(x0 + 1, WDIM - 1), y1 = min(y0 + 1, LDIM - 1);
            float lx = xc - (float)x0, hx = 1.0f - lx;
            float ly = yc - (float)y0, hy = 1.0f - ly;
            float vw = v ? 0.25f : 0.0f;   // xy validity + 2x2 xy-pool average
            sSpat[s4 * 4 + 0] = (x0 << 12) | (y0 << 6);  sW[s4 * 4 + 0] = vw * hx * hy;
            sSpat[s4 * 4 + 1] = (x1 << 12) | (y0 << 6);  sW[s4 * 4 + 1] = vw * lx * hy;
            sSpat[s4 * 4 + 2] = (x0 << 12) | (y1 << 6);  sW[s4 * 4 + 2] = vw * hx * ly;
            sSpat[s4 * 4 + 3] = (x1 << 12) | (y1 << 6);  sW[s4 * 4 + 3] = vw * lx * ly;
        }
    }
    __syncthreads();

    // ---- swizzle Zc into B-operand lane layout (ISA 7.12.2, 32-bit 4x16):
    // lane l, VGPR pair = { B[k0][l&15], B[k0+1][l&15] }, k0 = 4*ck + 2*(l>>4)
    for (int i = tid; i < 16 * 32; i += 256) {
        int ck = i >> 5, l = i & 31;
        int zb = 4 * ck + ((l >> 4) << 1);
        v2f bv; bv.x = sZc[l & 15][zb]; bv.y = sZc[l & 15][zb + 1];
        sB[ck][l] = bv;
    }
    __syncthreads();

    const int b     = sfirst(sMeta[0]);
    const int ckmin = sfirst(sMeta[1]);
    const int ckmax = sfirst(sMeta[2]);

    v8f acc = {0.f, 0.f, 0.f, 0.f, 0.f, 0.f, 0.f, 0.f};

    if (ckmin <= ckmax) {
        // A layout (16x4 f32): lane l -> M = l&15 (channel in group), K pair offset 2*(l>>4)
        const int ch = (wave << 4) + (lane & 15);
        const float* rowbase =
            in + (((size_t)(b * C_CH + ch)) << 18) + (size_t)((lane >> 4) << 1);

        for (int q = 0; q < 16; ++q) {
            float w = sfirst(sW[q]);          // uniform -> scalar branch keeps EXEC all-1s
            if (w == 0.0f) continue;
            const float* rp = rowbase + sfirst(sSpat[q]);
            for (int ck = ckmin; ck <= ckmax; ++ck) {
                v2f a = *(const v2f*)(rp + 4 * ck);   // two consecutive z values, 8B aligned
                a.x *= w; a.y *= w;                   // fold bilinear-xy weight into A
                v2f bb = sB[ck][lane];
                acc = __builtin_amdgcn_wmma_f32_16x16x4_f32(
                    false, a, false, bb, (short)0, acc, false, false);
            }
        }
    }

    // D layout: lane l holds N = l&15 (= z-bin), VGPR r holds M = r + 8*(l>>4)
    const int bz = lane & 15;
    if (bz < 7) {
        const int chbase = (wave << 4) + ((lane >> 4) << 3);
        const size_t obase = (size_t)roi * C_CH * 343 + (size_t)(bx * 49 + by * 7 + bz);
        #pragma unroll
        for (int r = 0; r < 8; ++r) {
            out[obase + (size_t)(chbase + r) * 343] = acc[r];
        }
    }
}

extern "C" void kernel_launch(void* const* d_in, const int* in_sizes, int n_in,
                              void* d_out, int out_size, void* d_ws, size_t ws_size,
                              hipStream_t stream) {
    const float* in    = (const float*)d_in[0];
    const float* rois  = (const float*)d_in[1];
    const float* scale = (const float*)d_in[2];
    float* out         = (float*)d_out;
    const int n_rois   = in_sizes[1] / 8;

    dim3 grid(49, (unsigned)n_rois);
    dim3 block(256);
    roi_align_rot3d_wmma<<<grid, block, 0, stream>>>(in, rois, scale, out);
}